// SimpleRNN_4389456576643
// MI455X (gfx1250) — compile-verified
//
#include <hip/hip_runtime.h>

#define VOCAB 50000
#define D 512
#define L 512
#define B 256
#define NC 20
#define NBLK 32  // one block per 16-wide column slice of D

typedef __attribute__((ext_vector_type(2))) float v2f;
typedef __attribute__((ext_vector_type(8))) float v8f;

__device__ __forceinline__ v8f wmma_f32_k4(v2f a, v2f b, v8f c) {
  // 8 args: (neg_a, A, neg_b, B, c_mod, C, reuse_a, reuse_b)
  return __builtin_amdgcn_wmma_f32_16x16x4_f32(false, a, false, b, (short)0, c,
                                               false, false);
}

// ---------------------------------------------------------------------------
// Phase 0: zero the grid-barrier counter (deterministic across graph replays)
// ---------------------------------------------------------------------------
__global__ void init_bar_kernel(unsigned int* bar) {
  if (threadIdx.x == 0) *bar = 0u;
}

// ---------------------------------------------------------------------------
// Phase 1: xin[l*B+b][:] = emb[x[b][l]] @ Wx + bias  (one wave = one 16x16 tile)
// rows r = l*B + b flattened; 16 consecutive rows share l (B % 16 == 0)
// ---------------------------------------------------------------------------
__global__ __launch_bounds__(256) void xin_proj_kernel(
    const int* __restrict__ x, const float* __restrict__ emb,
    const float* __restrict__ Wx, const float* __restrict__ bias,
    float* __restrict__ xin) {
  const int lane = threadIdx.x & 31;
  const int w = blockIdx.x * 8 + (threadIdx.x >> 5);  // global wave id
  const int tileRow = w >> 5;                         // (L*B)/16 row tiles
  const int tileCol = w & 31;                         // D/16 = 32 col tiles
  const int r0 = tileRow * 16;
  const int n0 = tileCol * 16;

  const int mLane = lane & 15;        // A: row in tile; B/C: col in tile
  const int kGrp  = (lane >> 4) * 2;  // K group 0 or 2

  // token for this lane's A-row
  const int row = r0 + mLane;
  const int bb  = row & (B - 1);
  const int ll  = row >> 8;  // row / B
  const int tok = x[bb * L + ll];
  const float* __restrict__ arow = emb + (long)tok * D;

  v8f c;
  const float bv = bias[n0 + mLane];
#pragma unroll
  for (int i = 0; i < 8; ++i) c[i] = bv;

#pragma unroll 4
  for (int kb = 0; kb < D; kb += 4) {
    v2f a, bm;
    a[0]  = arow[kb + kGrp + 0];
    a[1]  = arow[kb + kGrp + 1];
    bm[0] = Wx[(kb + kGrp + 0) * D + n0 + mLane];
    bm[1] = Wx[(kb + kGrp + 1) * D + n0 + mLane];
    c = wmma_f32_k4(a, bm, c);
  }

#pragma unroll
  for (int v = 0; v < 8; ++v) {
    const int m = (lane < 16) ? v : (v + 8);
    xin[(long)(r0 + m) * D + n0 + mLane] = c[v];
  }
}

// ---------------------------------------------------------------------------
// Phase 2: persistent cooperative scan: h_{t+1} = tanh(xin_t + h_t @ Wh)
// Block bid owns output columns [bid*16, bid*16+16): its Wh slice (512x16 fp32
// = 32 KB) is staged in LDS ONCE (K-pair interleaved for ds_load_b64) and
// reused across all 512 steps. 8 waves x 2 reps cover the 16 row tiles of B.
// Grid sync via monotonic atomic counter (reset by init kernel each launch).
// ---------------------------------------------------------------------------
__global__ __launch_bounds__(256) void rnn_scan_kernel(
    const float* __restrict__ xin, const float* __restrict__ Wh,
    float* __restrict__ hbuf, unsigned int* __restrict__ bar) {
  // LDS: Wh slice, layout lds_wh[(k>>1)*32 + n*2 + (k&1)], n = 0..15
  __shared__ float lds_wh[(D / 2) * 32];  // 8192 floats = 32 KB

  const int lane = threadIdx.x & 31;
  const int waveInBlk = threadIdx.x >> 5;
  const int mLane = lane & 15;
  const int kGrp  = (lane >> 4) * 2;
  const int n0    = blockIdx.x * 16;  // this block's column slice

  // cooperative load of Wh[:, n0:n0+16] into LDS (pair-interleaved)
  for (int idx = threadIdx.x; idx < D * 16; idx += 256) {
    const int k = idx >> 4;
    const int j = idx & 15;
    lds_wh[(k >> 1) * 32 + j * 2 + (k & 1)] = Wh[k * D + n0 + j];
  }
  __syncthreads();

  float* h0buf = hbuf;
  float* h1buf = hbuf + B * D;

  for (int t = 0; t < L; ++t) {
    const float* __restrict__ hc = (t & 1) ? h1buf : h0buf;
    float* __restrict__ hn = (t & 1) ? h0buf : h1buf;
    const float* __restrict__ xt = xin + (long)t * B * D;

#pragma unroll
    for (int rep = 0; rep < 2; ++rep) {
      const int m0 = (waveInBlk + rep * 8) * 16;  // row tile over B

      v8f c;
#pragma unroll
      for (int v = 0; v < 8; ++v) {
        const int m = (lane < 16) ? v : (v + 8);
        c[v] = xt[(m0 + m) * D + n0 + mLane];
      }

      if (t > 0) {  // h0 == 0, skip GEMM at step 0
        const float* __restrict__ arow = hc + (m0 + mLane) * D;
#pragma unroll 4
        for (int kb = 0; kb < D; kb += 4) {
          v2f a, bm;
          a[0] = arow[kb + kGrp + 0];
          a[1] = arow[kb + kGrp + 1];
          const float* __restrict__ bp =
              &lds_wh[((kb + kGrp) >> 1) * 32 + mLane * 2];
          bm[0] = bp[0];
          bm[1] = bp[1];
          c = wmma_f32_k4(a, bm, c);
        }
      }

#pragma unroll
      for (int v = 0; v < 8; ++v) {
        const int m = (lane < 16) ? v : (v + 8);
        hn[(m0 + m) * D + n0 + mLane] = tanhf(c[v]);
      }
    }

    // ---- grid-wide barrier ----
    __syncthreads();
    if (threadIdx.x == 0) {
      __threadfence();  // make this block's h_next visible device-wide
      atomicAdd(bar, 1u);
      const unsigned int target = (unsigned int)NBLK * (unsigned int)(t + 1);
      while (__hip_atomic_load(bar, __ATOMIC_ACQUIRE,
                               __HIP_MEMORY_SCOPE_AGENT) < target) {
        __builtin_amdgcn_s_sleep(2);
      }
    }
    __syncthreads();
  }
}

// ---------------------------------------------------------------------------
// Phase 3: logits = h_last @ Wd + bd ; softmax over NC=20 classes (tiny)
// ---------------------------------------------------------------------------
__global__ __launch_bounds__(128) void head_softmax_kernel(
    const float* __restrict__ h, const float* __restrict__ Wd,
    const float* __restrict__ bd, float* __restrict__ out) {
  const int b = blockIdx.x * blockDim.x + threadIdx.x;
  if (b >= B) return;
  float acc[NC];
#pragma unroll
  for (int c = 0; c < NC; ++c) acc[c] = bd[c];
  const float* __restrict__ hr = h + b * D;
  for (int k = 0; k < D; ++k) {
    const float hv = hr[k];
#pragma unroll
    for (int c = 0; c < NC; ++c) acc[c] = fmaf(hv, Wd[k * NC + c], acc[c]);
  }
  float mx = acc[0];
#pragma unroll
  for (int c = 1; c < NC; ++c) mx = fmaxf(mx, acc[c]);
  float s = 0.f;
#pragma unroll
  for (int c = 0; c < NC; ++c) {
    acc[c] = expf(acc[c] - mx);
    s += acc[c];
  }
  const float inv = 1.f / s;
#pragma unroll
  for (int c = 0; c < NC; ++c) out[b * NC + c] = acc[c] * inv;
}

// ---------------------------------------------------------------------------
extern "C" void kernel_launch(void* const* d_in, const int* in_sizes, int n_in,
                              void* d_out, int out_size, void* d_ws,
                              size_t ws_size, hipStream_t stream) {
  const int*   x    = (const int*)d_in[0];
  const float* emb  = (const float*)d_in[1];
  const float* Wx   = (const float*)d_in[2];
  const float* Wh   = (const float*)d_in[3];
  const float* bias = (const float*)d_in[4];
  const float* Wd   = (const float*)d_in[5];
  const float* bd   = (const float*)d_in[6];
  float* out = (float*)d_out;

  char* ws = (char*)d_ws;
  unsigned int* bar = (unsigned int*)ws;               // 256 B reserved
  float* hbuf = (float*)(ws + 256);                    // 2*B*D fp32 = 1 MiB
  float* xin  = (float*)(ws + 256 + (size_t)2 * B * D * sizeof(float));

  init_bar_kernel<<<1, 32, 0, stream>>>(bar);

  const int tiles1 = (L * B / 16) * (D / 16);          // 262144 tiles
  xin_proj_kernel<<<tiles1 / 8, 256, 0, stream>>>(x, emb, Wx, bias, xin);

  rnn_scan_kernel<<<NBLK, 256, 0, stream>>>(xin, Wh, hbuf, bar);

  // L=512 steps: step t writes buf[(t+1)&1]; t=511 -> buf0 == hbuf
  head_softmax_kernel<<<(B + 127) / 128, 128, 0, stream>>>(hbuf, Wd, bd, out);
}